// VectorQuantizer_36378372997743
// MI455X (gfx1250) — compile-verified
//
#include <hip/hip_runtime.h>
#include <hip/hip_bf16.h>

typedef __attribute__((ext_vector_type(2))) float v2f;
typedef __attribute__((ext_vector_type(8))) float v8f;

#define VOCAB 8192
#define ZC 32
#define NVEC 16384          // B*H*W = 16*32*32
#define OUT_ELEMS 524288    // 16*32*32*32
#define LDS_STRIDE 34       // pad 32-float rows -> conflict-free banks

// ---------------- Kernel 1: normalize codebook rows -------------------------
__global__ __launch_bounds__(256)
void vq_norm_emb(const float* __restrict__ emb, float* __restrict__ en) {
    int v = blockIdx.x * 256 + threadIdx.x;
    if (v >= VOCAB) return;
    const float* src = emb + v * ZC;
    float ss = 0.f;
#pragma unroll
    for (int c = 0; c < ZC; ++c) { float x = src[c]; ss += x * x; }
    float inv = 1.f / fmaxf(sqrtf(ss), 1e-12f);
    float* dst = en + v * ZC;
#pragma unroll
    for (int c = 0; c < ZC; ++c) dst[c] = src[c] * inv;
}

// ---------------- Kernel 2: permute [B,C,H,W]->[N,C] and normalize ----------
__global__ __launch_bounds__(256)
void vq_norm_z(const float* __restrict__ z, float* __restrict__ zn) {
    int n = blockIdx.x * 256 + threadIdx.x;   // n = b*1024 + h*32 + w
    if (n >= NVEC) return;
    int b = n >> 10, hw = n & 1023;
    float v[ZC];
    float ss = 0.f;
#pragma unroll
    for (int c = 0; c < ZC; ++c) {
        float x = z[b * (ZC * 1024) + c * 1024 + hw];   // coalesced per c
        v[c] = x; ss += x * x;
    }
    float inv = 1.f / fmaxf(sqrtf(ss), 1e-12f);
    float* dst = zn + n * ZC;
#pragma unroll
    for (int c = 0; c < ZC; ++c) dst[c] = v[c] * inv;
}

// ---------------- Kernel 3: WMMA f32 scores + fused argmax ------------------
// 8 waves/block; each wave owns 16 query rows; block stages 128 codes in LDS.
__global__ __launch_bounds__(256)
void vq_wmma_argmax(const float* __restrict__ zn, const float* __restrict__ en,
                    int* __restrict__ idxOut) {
    __shared__ float smem[128 * LDS_STRIDE];

    const int tid  = threadIdx.x;
    const int wave = tid >> 5;
    const int lane = tid & 31;
    const int r    = lane & 15;   // row (A) / column (B) within tile
    const int h    = lane >> 4;   // lane-half selects K-pair (ISA 32-bit layout)
    const int rowBase = blockIdx.x * 128 + wave * 16;
    const int row = rowBase + r;

    // A operand: 16x4 f32 per step; lane r holds M=r, VGPR0/1 = K = 2h / 2h+1.
    v2f a[8];
#pragma unroll
    for (int k = 0; k < 8; ++k)
        a[k] = *reinterpret_cast<const v2f*>(zn + row * ZC + 4 * k + 2 * h);

    float best[8];
    int   bidx[8];
#pragma unroll
    for (int j = 0; j < 8; ++j) { best[j] = -3.0e38f; bidx[j] = 0; }

    for (int cb = 0; cb < VOCAB; cb += 128) {
        // cooperative stage of 128 codes (4096 floats) into padded LDS
#pragma unroll
        for (int i = 0; i < 16; ++i) {
            int fid  = tid + 256 * i;          // 0..4095, coalesced
            int code = fid >> 5, c = fid & 31;
            smem[code * LDS_STRIDE + c] = en[(cb + code) * ZC + c];
        }
        __syncthreads();
        if (cb + 128 < VOCAB)                  // global_prefetch_b8 of next chunk
            __builtin_prefetch(&en[(cb + 128) * ZC + tid * 16], 0, 1);

#pragma unroll
        for (int sub = 0; sub < 8; ++sub) {
            const int lc = sub * 16 + r;       // B column N=r
            v8f acc = {};
#pragma unroll
            for (int k = 0; k < 8; ++k) {
                v2f b = *reinterpret_cast<const v2f*>(
                    &smem[lc * LDS_STRIDE + 4 * k + 2 * h]);
                acc = __builtin_amdgcn_wmma_f32_16x16x4_f32(
                    false, a[k], false, b, (short)0, acc, false, false);
            }
            const int code = cb + sub * 16 + r;
            // C/D layout: lane half 0 -> rows j (0..7); half 1 -> rows j+8.
#pragma unroll
            for (int j = 0; j < 8; ++j) {
                float s = acc[j];
                if (s > best[j]) { best[j] = s; bidx[j] = code; }
            }
        }
        __syncthreads();
    }

    // reduce across the 16 lanes of each half (xor masks stay within a half)
#pragma unroll
    for (int off = 1; off < 16; off <<= 1) {
#pragma unroll
        for (int j = 0; j < 8; ++j) {
            float ov = __shfl_xor(best[j], off, 32);
            int   oi = __shfl_xor(bidx[j], off, 32);
            if (ov > best[j] || (ov == best[j] && oi < bidx[j])) {
                best[j] = ov; bidx[j] = oi;
            }
        }
    }

    // lane j (half 0) writes row rowBase+j; lane 16+j writes row rowBase+8+j
#pragma unroll
    for (int j = 0; j < 8; ++j) {
        if (lane == j)      idxOut[rowBase + j]     = bidx[j];
        if (lane == 16 + j) idxOut[rowBase + 8 + j] = bidx[j];
    }
}

// ---------------- Kernel 4: gather + permute out + partial loss sums --------
__global__ __launch_bounds__(256)
void vq_gather_out(const float* __restrict__ en, const float* __restrict__ zn,
                   const int* __restrict__ idx, float* __restrict__ out,
                   float* __restrict__ partials) {
    __shared__ float red[256];
    int n = blockIdx.x * 256 + threadIdx.x;
    int b = n >> 10, hw = n & 1023;
    int code = idx[n];
    float s = 0.f;
#pragma unroll
    for (int c = 0; c < ZC; ++c) {
        float q = en[code * ZC + c];                 // z_q = l2norm(emb[idx])
        float d = q - zn[n * ZC + c];
        s += d * d;
        out[b * (ZC * 1024) + c * 1024 + hw] = q;    // [B,C,H,W], coalesced per c
    }
    red[threadIdx.x] = s;
    __syncthreads();
    for (int off = 128; off > 0; off >>= 1) {
        if (threadIdx.x < off) red[threadIdx.x] += red[threadIdx.x + off];
        __syncthreads();
    }
    if (threadIdx.x == 0) partials[blockIdx.x] = red[0];
}

// ---------------- Kernel 5: deterministic final reduction -------------------
__global__ void vq_finalize(const float* __restrict__ partials,
                            float* __restrict__ scalars) {
    if (threadIdx.x == 0) {
        float s = 0.f;
        for (int i = 0; i < 64; ++i) s += partials[i];
        float m = s / (float)OUT_ELEMS;
        scalars[0] = m;          // vq_loss
        scalars[1] = 0.25f * m;  // commit_loss = BETA * mean
    }
}

extern "C" void kernel_launch(void* const* d_in, const int* in_sizes, int n_in,
                              void* d_out, int out_size, void* d_ws, size_t ws_size,
                              hipStream_t stream) {
    const float* z   = (const float*)d_in[0];   // [16,32,32,32]
    const float* emb = (const float*)d_in[1];   // [8192,32]
    float* out = (float*)d_out;                 // z_q (524288) + vq + commit

    float* ws = (float*)d_ws;
    float* zn       = ws;                        // 524288 f32
    float* en       = ws + OUT_ELEMS;            // 262144 f32
    int*   idx      = (int*)(ws + OUT_ELEMS + VOCAB * ZC);   // 16384 i32
    float* partials = ws + OUT_ELEMS + VOCAB * ZC + NVEC;    // 64 f32

    vq_norm_emb<<<VOCAB / 256, 256, 0, stream>>>(emb, en);
    vq_norm_z<<<NVEC / 256, 256, 0, stream>>>(z, zn);
    vq_wmma_argmax<<<NVEC / 128, 256, 0, stream>>>(zn, en, idx);
    vq_gather_out<<<NVEC / 256, 256, 0, stream>>>(en, zn, idx, out, partials);
    vq_finalize<<<1, 32, 0, stream>>>(partials, out + OUT_ELEMS);
}